// Kmeans_47029891891617
// MI455X (gfx1250) — compile-verified
//
#include <hip/hip_runtime.h>

typedef __attribute__((ext_vector_type(2))) float v2f;
typedef __attribute__((ext_vector_type(8))) float v8f;

#define KC 64         // number of centers
#define NG 4          // KC/16 center groups
#define MAX_ITER 5    // reference fixes max_iter = 5

// Workspace layout (floats):
//  [0, 256)    gacc[64][4]   : (sum_x, sum_y, sum_z, count) global accumulators
//  [256, 448)  centers[64][3]
//  [448, 704)  Amat[4][32][2]: per-lane WMMA A operand (centers), group-major
#define WS_ACC 0
#define WS_CTR 256
#define WS_A   448

// ---------------------------------------------------------------------------
// Init: zero accumulators, centers = centers0, build A operand in wave layout.
// A row for center c is (-2cx, -2cy, -2cz, ||c||^2); lanes 0-15 hold (K0,K1),
// lanes 16-31 hold (K2,K3) per the 32-bit 16x4 A-operand layout.
// ---------------------------------------------------------------------------
__global__ void km_init(const float* __restrict__ c0, float* __restrict__ ws) {
    int t = threadIdx.x;              // blockDim = 256
    ws[WS_ACC + t] = 0.0f;
    if (t < KC * 3) ws[WS_CTR + t] = c0[t];
    if (t < NG * 32) {
        int g = t >> 5, L = t & 31, c = g * 16 + (L & 15);
        float cx = c0[c * 3 + 0], cy = c0[c * 3 + 1], cz = c0[c * 3 + 2];
        float a0, a1;
        if (L < 16) { a0 = -2.0f * cx; a1 = -2.0f * cy; }
        else        { a0 = -2.0f * cz; a1 = cx * cx + cy * cy + cz * cz; }
        ws[WS_A + (g * 32 + L) * 2 + 0] = a0;
        ws[WS_A + (g * 32 + L) * 2 + 1] = a1;
    }
}

// ---------------------------------------------------------------------------
// Assign + accumulate: one streaming pass per iteration.
// D = centers(16x4) x points(4x16) via V_WMMA_F32_16X16X4_F32, so each lane
// holds 8 scores (one point, 8 centers) per WMMA; the 64-way argmin is a
// 32-candidate register reduction plus one half-wave shuffle merge.
// ---------------------------------------------------------------------------
__global__ void km_assign(const float* __restrict__ data,
                          float* __restrict__ ws, int n) {
    __shared__ float lacc[KC * 4];
    int t = threadIdx.x;              // blockDim = 256 == KC*4
    lacc[t] = 0.0f;
    __syncthreads();

    int lane  = t & 31;
    bool lo   = lane < 16;
    int half8 = lo ? 0 : 8;

    // Per-lane A operands (centers) for the 4 groups (coalesced, L2-hot).
    const float* Am = ws + WS_A;
    v2f a[NG];
#pragma unroll
    for (int g = 0; g < NG; ++g) {
        a[g].x = Am[(g * 32 + lane) * 2 + 0];
        a[g].y = Am[(g * 32 + lane) * 2 + 1];
    }

    int npairs      = n >> 5;                       // 32 points per loop trip
    int waves_total = gridDim.x * (blockDim.x >> 5);
    int gw          = blockIdx.x * (blockDim.x >> 5) + (t >> 5);

    for (int pr = gw; pr < npairs; pr += waves_total) {
        // lane<16 owns tile-0 point p = pr*32+lane; lane>=16 owns tile-1
        // point p = pr*32+lane (i.e. pr*32+16+(lane-16)). No duplicate loads.
        int p = pr * 32 + lane;
        const float* dp = data + 3 * p;
        float x = dp[0], y = dp[1], z = dp[2];
        __builtin_prefetch(data + 3 * (p + 32 * waves_total), 0, 1);

        // Build both 4x16 point operands via half-wave swaps:
        //   b0 (tile 0): lanes<16 -> (x,y) own; lanes>=16 -> (z of lane^16, 1)
        //   b1 (tile 1): lanes<16 -> (x,y of lane^16);  lanes>=16 -> (z own, 1)
        float xs = __shfl_xor(x, 16, 32);
        float ys = __shfl_xor(y, 16, 32);
        float zs = __shfl_xor(z, 16, 32);
        v2f b0, b1;
        b0.x = lo ? x  : zs;  b0.y = lo ? y  : 1.0f;
        b1.x = lo ? xs : z;   b1.y = lo ? ys : 1.0f;

        v8f z8 = {};
        v8f d0[NG], d1[NG];
#pragma unroll
        for (int g = 0; g < NG; ++g) {
            d0[g] = __builtin_amdgcn_wmma_f32_16x16x4_f32(false, a[g], false, b0,
                                                          (short)0, z8, false, false);
            d1[g] = __builtin_amdgcn_wmma_f32_16x16x4_f32(false, a[g], false, b1,
                                                          (short)0, z8, false, false);
        }

        // Branch-free 32-candidate register argmin per tile (ids increasing,
        // strict < keeps the lowest index on ties).
        float bv0 = d0[0][0], bv1 = d1[0][0];
        int   bi0 = half8,    bi1 = half8;
#pragma unroll
        for (int g = 0; g < NG; ++g) {
#pragma unroll
            for (int r = 0; r < 8; ++r) {
                if (g == 0 && r == 0) continue;
                int id = g * 16 + half8 + r;
                float v0 = d0[g][r];
                bool c0 = v0 < bv0;
                bv0 = c0 ? v0 : bv0;
                bi0 = c0 ? id : bi0;
                float v1 = d1[g][r];
                bool c1 = v1 < bv1;
                bv1 = c1 ? v1 : bv1;
                bi1 = c1 ? id : bi1;
            }
        }
        // Merge the two center-halves (lane^16 holds the complementary 32).
        {
            float ov = __shfl_xor(bv0, 16, 32);
            int   oi = __shfl_xor(bi0, 16, 32);
            bool c = (ov < bv0) | ((ov == bv0) & (oi < bi0));
            bi0 = c ? oi : bi0;
        }
        {
            float ov = __shfl_xor(bv1, 16, 32);
            int   oi = __shfl_xor(bi1, 16, 32);
            bool c = (ov < bv1) | ((ov == bv1) & (oi < bi1));
            bi1 = c ? oi : bi1;
        }

        // Every lane accumulates exactly its own point -> zero divergence.
        int bidx = lo ? bi0 : bi1;
        atomicAdd(&lacc[bidx * 4 + 0], x);
        atomicAdd(&lacc[bidx * 4 + 1], y);
        atomicAdd(&lacc[bidx * 4 + 2], z);
        atomicAdd(&lacc[bidx * 4 + 3], 1.0f);
    }

    __syncthreads();
    atomicAdd(&ws[WS_ACC + t], lacc[t]);   // one global atomic per thread
}

// ---------------------------------------------------------------------------
// Update: centers = sums / max(count,1); rebuild A operand; re-zero acc.
// Also writes centers to d_out (final iteration's values persist).
// ---------------------------------------------------------------------------
__global__ void km_update(float* __restrict__ ws, float* __restrict__ out) {
    __shared__ float cs[KC * 3];
    int t = threadIdx.x;              // blockDim = 256
    if (t < KC) {
        float cnt = ws[WS_ACC + t * 4 + 3];
        float denom = fmaxf(cnt, 1.0f);
        float cx = ws[WS_ACC + t * 4 + 0] / denom;
        float cy = ws[WS_ACC + t * 4 + 1] / denom;
        float cz = ws[WS_ACC + t * 4 + 2] / denom;
        ws[WS_CTR + t * 3 + 0] = cx; out[t * 3 + 0] = cx; cs[t * 3 + 0] = cx;
        ws[WS_CTR + t * 3 + 1] = cy; out[t * 3 + 1] = cy; cs[t * 3 + 1] = cy;
        ws[WS_CTR + t * 3 + 2] = cz; out[t * 3 + 2] = cz; cs[t * 3 + 2] = cz;
    }
    __syncthreads();
    ws[WS_ACC + t] = 0.0f;            // ready for next iteration
    if (t < NG * 32) {
        int g = t >> 5, L = t & 31, c = g * 16 + (L & 15);
        float cx = cs[c * 3 + 0], cy = cs[c * 3 + 1], cz = cs[c * 3 + 2];
        float a0, a1;
        if (L < 16) { a0 = -2.0f * cx; a1 = -2.0f * cy; }
        else        { a0 = -2.0f * cz; a1 = cx * cx + cy * cy + cz * cz; }
        ws[WS_A + (g * 32 + L) * 2 + 0] = a0;
        ws[WS_A + (g * 32 + L) * 2 + 1] = a1;
    }
}

extern "C" void kernel_launch(void* const* d_in, const int* in_sizes, int n_in,
                              void* d_out, int out_size, void* d_ws, size_t ws_size,
                              hipStream_t stream) {
    const float* data = (const float*)d_in[0];
    const float* c0   = (const float*)d_in[1];
    float* ws  = (float*)d_ws;
    float* out = (float*)d_out;
    int n = in_sizes[0] / 3;

    km_init<<<1, 256, 0, stream>>>(c0, ws);
    for (int it = 0; it < MAX_ITER; ++it) {
        km_assign<<<1024, 256, 0, stream>>>(data, ws, n);
        km_update<<<1, 256, 0, stream>>>(ws, out);
    }
}